// ExpectedCalibrationError_40063454937729
// MI455X (gfx1250) — compile-verified
//
#include <hip/hip_runtime.h>

// ECE: N x 128 float32 probs, int32 targets -> scalar float.
// HBM-bound: 1.024 GB read once -> ~44us floor at 23.3 TB/s. No GEMM -> WMMA N/A.
// CDNA5 path: per-wave double-buffered GLOBAL_LOAD_ASYNC_TO_LDS_B128 staging
// (ASYNCcnt / s_wait_asynccnt), lane-owns-row LDS scan (528B padded rows ->
// conflict-free ds_load_b128), per-thread LDS bin slots (no atomics, deterministic),
// block partials -> d_ws, fixed-order finalize kernel.

#define NUM_BINS 10
#define THREADS 128
#define WAVES_PER_BLOCK (THREADS / 32)
#define TILE_ROWS 32
#define C_DIM 128
#define ROW_BYTES (C_DIM * 4)                                  // 512
#define LDS_ROW_STRIDE (ROW_BYTES + 16)                        // 528: kills 64-bank conflicts
#define TILE_LDS_BYTES (TILE_ROWS * LDS_ROW_STRIDE)            // 16896
#define NBUF 2
#define TILES_LDS_TOTAL (WAVES_PER_BLOCK * NBUF * TILE_LDS_BYTES)  // 135168
#define BINS_LDS_BYTES (THREADS * 2 * NUM_BINS * 4)            // 10240
#define SMEM_BYTES (TILES_LDS_TOTAL + BINS_LDS_BYTES)          // 145408 (< 320KB WGP LDS)
#define MAX_BLOCKS 1024

extern __shared__ char smem[];

// Issue one tile: 32 async b128 copies, each moving 512B (row I) coalesced from
// global into the padded LDS row. VDST VGPR = LDS byte offset, VADDR = 32-bit
// global byte offset (GVS mode, saddr = tensor base).
template <int I>
__device__ __forceinline__ void issue_tile_copy(unsigned lds_off, int voff,
                                                const float* __restrict__ base) {
  if constexpr (I < TILE_ROWS) {
    asm volatile("global_load_async_to_lds_b128 %0, %1, %2"
                 :
                 : "v"(lds_off + I * LDS_ROW_STRIDE),
                   "v"(voff + I * ROW_BYTES),
                   "s"(base)
                 : "memory");
    issue_tile_copy<I + 1>(lds_off, voff, base);
  }
}

// Scan one 128-float row (16B chunks), strict '>' keeps first-occurrence argmax
// exactly like jnp.argmax.
__device__ __forceinline__ void scan_row(const float4* __restrict__ row,
                                         float& lm, int& li) {
  lm = -1.0f;
  li = 0;
#pragma unroll
  for (int j = 0; j < C_DIM / 4; ++j) {
    float4 q = row[j];
    int b = j * 4;
    if (q.x > lm) { lm = q.x; li = b; }
    if (q.y > lm) { lm = q.y; li = b + 1; }
    if (q.z > lm) { lm = q.z; li = b + 2; }
    if (q.w > lm) { lm = q.w; li = b + 3; }
  }
}

__device__ __forceinline__ void bin_update(float* __restrict__ mybins,
                                           float lm, int li, int tgt) {
  // searchsorted(linspace(0,1,11), conf, 'right') - 1 == clamp(floor(10*conf))
  int bin = (int)(lm * (float)NUM_BINS);
  bin = bin < 0 ? 0 : (bin > NUM_BINS - 1 ? NUM_BINS - 1 : bin);
  mybins[bin] += lm;
  mybins[NUM_BINS + bin] += (li == tgt) ? 1.0f : 0.0f;
}

__global__ __launch_bounds__(THREADS)
void ece_tiles_kernel(const float* __restrict__ inputs,
                      const int* __restrict__ targets,
                      float* __restrict__ partials,
                      int n_rows) {
  const int tid = threadIdx.x;
  const int lane = tid & 31;
  const int wave = tid >> 5;

  // Per-thread private 20-float bin slot in LDS: race-free, deterministic.
  float* mybins = (float*)(smem + TILES_LDS_TOTAL) + tid * 2 * NUM_BINS;
#pragma unroll
  for (int b = 0; b < 2 * NUM_BINS; ++b) mybins[b] = 0.0f;

  const long long T = (long long)n_rows / TILE_ROWS;
  const long long gwave = (long long)blockIdx.x * WAVES_PER_BLOCK + wave;
  const long long nwaves = (long long)gridDim.x * WAVES_PER_BLOCK;

  // Low 32 bits of a generic LDS pointer == wave-relative LDS byte offset.
  const unsigned smem_lds = (unsigned)(unsigned long long)(void*)smem;
  const unsigned obuf0 = (unsigned)(wave * NBUF * TILE_LDS_BYTES);
  const unsigned obuf[2] = {obuf0, obuf0 + TILE_LDS_BYTES};

  long long tile = gwave;
  int p = 0;
  if (tile < T) {
    issue_tile_copy<0>(smem_lds + obuf[0] + (unsigned)lane * 16,
                       (int)(tile * (long long)TILE_ROWS * ROW_BYTES) + lane * 16,
                       inputs);
  }
  for (; tile < T; tile += nwaves) {
    const long long nxt = tile + nwaves;
    if (nxt < T) {
      issue_tile_copy<0>(smem_lds + obuf[p ^ 1] + (unsigned)lane * 16,
                         (int)(nxt * (long long)TILE_ROWS * ROW_BYTES) + lane * 16,
                         inputs);
      // 32 older (current tile) + 32 newer (next tile) outstanding; async loads
      // complete in order, so <=32 means the current tile is resident.
      asm volatile("s_wait_asynccnt 32" ::: "memory");
    } else {
      asm volatile("s_wait_asynccnt 0" ::: "memory");
    }

    // Lane l owns row l of the tile: 32x ds_load_b128, 2-way banked (528B stride).
    const float4* row =
        (const float4*)(smem + obuf[p] + (unsigned)lane * LDS_ROW_STRIDE);
    float lm;
    int li;
    scan_row(row, lm, li);
    int tgt = targets[tile * TILE_ROWS + lane];  // coalesced b32
    bin_update(mybins, lm, li, tgt);
    p ^= 1;
  }

  // Tail rows (n_rows % 32), handled by global wave 0 straight from HBM.
  const long long tail0 = T * TILE_ROWS;
  if (gwave == 0 && tail0 + lane < (long long)n_rows) {
    const float4* row = (const float4*)(inputs + (size_t)(tail0 + lane) * C_DIM);
    float lm;
    int li;
    scan_row(row, lm, li);
    int tgt = targets[tail0 + lane];
    bin_update(mybins, lm, li, tgt);
  }

  __syncthreads();
  if (tid < 2 * NUM_BINS) {
    const float* bb = (const float*)(smem + TILES_LDS_TOTAL);
    float s = 0.0f;
    for (int t = 0; t < THREADS; ++t) s += bb[t * 2 * NUM_BINS + tid];
    partials[(size_t)blockIdx.x * 2 * NUM_BINS + tid] = s;
  }
}

__global__ __launch_bounds__(THREADS)
void ece_finalize_kernel(const float* __restrict__ partials,
                         int nblocks,
                         float* __restrict__ out,
                         float inv_n) {
  __shared__ float s[THREADS];
  __shared__ float s_final[2 * NUM_BINS];
  float acc[2 * NUM_BINS];
#pragma unroll
  for (int b = 0; b < 2 * NUM_BINS; ++b) acc[b] = 0.0f;

  for (int j = threadIdx.x; j < nblocks; j += THREADS) {
#pragma unroll
    for (int b = 0; b < 2 * NUM_BINS; ++b)
      acc[b] += partials[(size_t)j * 2 * NUM_BINS + b];
  }

  for (int b = 0; b < 2 * NUM_BINS; ++b) {
    s[threadIdx.x] = acc[b];
    __syncthreads();
    for (int o = THREADS / 2; o > 0; o >>= 1) {
      if (threadIdx.x < o) s[threadIdx.x] += s[threadIdx.x + o];
      __syncthreads();
    }
    if (threadIdx.x == 0) s_final[b] = s[0];
    __syncthreads();
  }

  if (threadIdx.x == 0) {
    float e = 0.0f;
#pragma unroll
    for (int b = 0; b < NUM_BINS; ++b)
      e += fabsf(s_final[NUM_BINS + b] - s_final[b]);
    out[0] = e * inv_n;
  }
}

extern "C" void kernel_launch(void* const* d_in, const int* in_sizes, int n_in,
                              void* d_out, int out_size, void* d_ws, size_t ws_size,
                              hipStream_t stream) {
  const float* inputs = (const float*)d_in[0];  // (N, 128) float32
  const int* targets = (const int*)d_in[1];     // (N,) int32
  float* out = (float*)d_out;                   // scalar float32
  float* partials = (float*)d_ws;               // [blocks][2*NUM_BINS]

  const int n_rows = in_sizes[1];

  int blocks = (int)(ws_size / (2 * NUM_BINS * sizeof(float)));
  if (blocks > MAX_BLOCKS) blocks = MAX_BLOCKS;
  if (blocks < 1) blocks = 1;

  ece_tiles_kernel<<<blocks, THREADS, SMEM_BYTES, stream>>>(inputs, targets,
                                                            partials, n_rows);
  ece_finalize_kernel<<<1, THREADS, 0, stream>>>(partials, blocks, out,
                                                 1.0f / (float)n_rows);
}